// Quantizer_44650480009908
// MI455X (gfx1250) — compile-verified
//
#include <hip/hip_runtime.h>

typedef __attribute__((ext_vector_type(2))) float v2f;
typedef __attribute__((ext_vector_type(8))) float v8f;
typedef int v4i __attribute__((vector_size(16)));

#define D_MODEL    1024
#define CODE_DEPTH 64
#define NUM_CODES  8192
#define NUM_ROWS   16384   // 8 * 2048

#define TILE_N  16         // codes per staged tile
#define ROW_PAD 68         // 64 floats + 4 pad -> rotates LDS banks per row

#define NSPLIT  8                            // codebook splits (parallelism)
#define CODES_PER_SPLIT (NUM_CODES / NSPLIT) // 1024 codes per split

#if defined(__gfx1250__) && __has_builtin(__builtin_amdgcn_global_load_async_to_lds_b128)
#define HAVE_ASYNC_LDS 1
typedef __attribute__((address_space(1))) v4i* gv4i_p;   // global src
typedef __attribute__((address_space(3))) v4i* lv4i_p;   // LDS dst
#endif

// Stage 16 bytes global -> LDS. Async (ASYNCcnt-tracked) when available.
__device__ __forceinline__ void stage16(const float* g, float* l) {
#ifdef HAVE_ASYNC_LDS
  __builtin_amdgcn_global_load_async_to_lds_b128(
      (gv4i_p)g, (lv4i_p)l, /*offset=*/0, /*cpol=*/0);
#else
  *(float4*)l = *(const float4*)g;
#endif
}

__device__ __forceinline__ void stage_wait() {
#ifdef HAVE_ASYNC_LDS
#if __has_builtin(__builtin_amdgcn_s_wait_asynccnt)
  __builtin_amdgcn_s_wait_asynccnt(0);
#else
  asm volatile("s_wait_asynccnt 0" ::: "memory");
#endif
#endif
}

// ---------------------------------------------------------------------------
// Kernel A: soft = x @ W^T + b   (M=16384, N=64, K=1024), f32 WMMA 16x16x4
// ---------------------------------------------------------------------------
__global__ __launch_bounds__(256) void k_down(const float* __restrict__ x,
                                              const float* __restrict__ W,
                                              const float* __restrict__ bias,
                                              float* __restrict__ soft) {
  const int lane = threadIdx.x & 31;
  const int wave = threadIdx.x >> 5;
  const int l15  = lane & 15;
  const int half = lane >> 4;
  const int m0   = (blockIdx.x * 8 + wave) * 16;

  v8f acc0 = {}, acc1 = {}, acc2 = {}, acc3 = {};

  const float* xrow = x + (size_t)(m0 + l15) * D_MODEL + 2 * half;
  const float* w0   = W + (size_t)( 0 + l15) * D_MODEL + 2 * half;
  const float* w1   = W + (size_t)(16 + l15) * D_MODEL + 2 * half;
  const float* w2   = W + (size_t)(32 + l15) * D_MODEL + 2 * half;
  const float* w3   = W + (size_t)(48 + l15) * D_MODEL + 2 * half;

#pragma unroll 4
  for (int k0 = 0; k0 < D_MODEL; k0 += 4) {
    v2f a  = *(const v2f*)(xrow + k0);
    v2f b0 = *(const v2f*)(w0 + k0);
    v2f b1 = *(const v2f*)(w1 + k0);
    v2f b2 = *(const v2f*)(w2 + k0);
    v2f b3 = *(const v2f*)(w3 + k0);
    acc0 = __builtin_amdgcn_wmma_f32_16x16x4_f32(false, a, false, b0, (short)0, acc0, false, false);
    acc1 = __builtin_amdgcn_wmma_f32_16x16x4_f32(false, a, false, b1, (short)0, acc1, false, false);
    acc2 = __builtin_amdgcn_wmma_f32_16x16x4_f32(false, a, false, b2, (short)0, acc2, false, false);
    acc3 = __builtin_amdgcn_wmma_f32_16x16x4_f32(false, a, false, b3, (short)0, acc3, false, false);
  }

  v8f accs[4] = {acc0, acc1, acc2, acc3};
#pragma unroll
  for (int t = 0; t < 4; ++t) {
    const int n = t * 16 + l15;
    const float bn = bias[n];
#pragma unroll
    for (int j = 0; j < 8; ++j) {
      const int m = m0 + j + 8 * half;
      soft[(size_t)m * CODE_DEPTH + n] = accs[t][j] + bn;
    }
  }
}

// ---------------------------------------------------------------------------
// c2[k] = sum_c book[k][c]^2
// ---------------------------------------------------------------------------
__global__ __launch_bounds__(256) void k_c2(const float* __restrict__ book,
                                            float* __restrict__ c2) {
  const int n = blockIdx.x * blockDim.x + threadIdx.x;
  if (n < NUM_CODES) {
    float s = 0.f;
#pragma unroll 8
    for (int c = 0; c < CODE_DEPTH; ++c) {
      const float v = book[(size_t)n * CODE_DEPTH + c];
      s += v * v;
    }
    c2[n] = s;
  }
}

// ---------------------------------------------------------------------------
// Kernel B: partial argmin over a 1/NSPLIT slice of codes of
// c2[n] - 2 * (soft @ book^T).  Grid = 64 row-blocks x NSPLIT code-splits
// = 512 workgroups (fills the WGP array).  8 waves/block, 32 rows/wave
// (2 A-fragment sets share each B fragment), 256 rows/block.  Codebook
// tiles (16 codes x 64 f32 = 4KB) double-buffered through LDS with async
// global->LDS copies overlapping the WMMA work on the current tile.
// ---------------------------------------------------------------------------
__global__ __launch_bounds__(256) void k_argmin(const float* __restrict__ soft,
                                                const float* __restrict__ book,
                                                const float* __restrict__ c2,
                                                float* __restrict__ bdp,
                                                int* __restrict__ bip) {
  __shared__ float tile[2][TILE_N * ROW_PAD];

  const int split = blockIdx.x & (NSPLIT - 1);
  const int rb    = blockIdx.x / NSPLIT;
  const int nbase = split * CODES_PER_SPLIT;

  const int lane = threadIdx.x & 31;
  const int wave = threadIdx.x >> 5;
  const int l15  = lane & 15;
  const int half = lane >> 4;
  const int m0   = (rb * 8 + wave) * 32;

  // Two A-fragment sets: rows m0..m0+15 and m0+16..m0+31, full K = 64.
  v2f a0[16], a1[16];
  {
    const float* s0 = soft + (size_t)(m0 + l15) * CODE_DEPTH + 2 * half;
    const float* s1 = soft + (size_t)(m0 + 16 + l15) * CODE_DEPTH + 2 * half;
#pragma unroll
    for (int kk = 0; kk < 16; ++kk) {
      a0[kk] = *(const v2f*)(s0 + kk * 4);
      a1[kk] = *(const v2f*)(s1 + kk * 4);
    }
  }

  // Staging assignment: thread -> one b128 (16B) of the 4KB tile.
  const int srow = threadIdx.x >> 4;          // 0..15 (code within tile)
  const int scol = (threadIdx.x & 15) << 2;   // 0,4,..,60 (float col)
  const float* gsrc = book + ((size_t)nbase + srow) * CODE_DEPTH + scol;
  float* ldst0 = &tile[0][srow * ROW_PAD + scol];
  float* ldst1 = &tile[1][srow * ROW_PAD + scol];

  // Preload tile 0.
  stage16(gsrc, ldst0);
  stage_wait();
  __syncthreads();

  float bd0[8], bd1[8];
  int   bi0[8], bi1[8];
#pragma unroll
  for (int j = 0; j < 8; ++j) {
    bd0[j] = 3.4e38f; bd1[j] = 3.4e38f; bi0[j] = 0; bi1[j] = 0;
  }

  int cur = 0;
  for (int t = 0; t < CODES_PER_SPLIT / TILE_N; ++t) {
    const int n0 = nbase + t * TILE_N;
    // Prefetch next tile into the other buffer (disjoint from readers).
    if (t + 1 < CODES_PER_SPLIT / TILE_N)
      stage16(gsrc + (size_t)(t + 1) * TILE_N * CODE_DEPTH,
              (cur ? ldst0 : ldst1));

    const float* bt = &tile[cur][l15 * ROW_PAD + 2 * half];
    v8f accA = {}, accB = {};
#pragma unroll
    for (int kk = 0; kk < 16; ++kk) {
      v2f bb = *(const v2f*)(bt + kk * 4);   // ds_load_b64, conflict-free
      accA = __builtin_amdgcn_wmma_f32_16x16x4_f32(false, a0[kk], false, bb, (short)0, accA, false, false);
      accB = __builtin_amdgcn_wmma_f32_16x16x4_f32(false, a1[kk], false, bb, (short)0, accB, false, false);
    }

    const int   n   = n0 + l15;      // this lane's code column (absolute)
    const float c2n = c2[n];
#pragma unroll
    for (int j = 0; j < 8; ++j) {
      const float dA = c2n - 2.0f * accA[j];
      const float dB = c2n - 2.0f * accB[j];
      if (dA < bd0[j]) { bd0[j] = dA; bi0[j] = n; }
      if (dB < bd1[j]) { bd1[j] = dB; bi1[j] = n; }
    }

    stage_wait();       // my async writes of tile t+1 complete
    __syncthreads();    // all waves: writes done + reads of tile t done
    cur ^= 1;
  }

  // Cross-lane argmin within each 16-lane half; ties -> smallest index.
#pragma unroll
  for (int j = 0; j < 8; ++j) {
    float bdA = bd0[j]; int biA = bi0[j];
    float bdB = bd1[j]; int biB = bi1[j];
#pragma unroll
    for (int off = 1; off < 16; off <<= 1) {
      const float odA = __shfl_xor(bdA, off, 16);
      const int   oiA = __shfl_xor(biA, off, 16);
      const float odB = __shfl_xor(bdB, off, 16);
      const int   oiB = __shfl_xor(biB, off, 16);
      if (odA < bdA || (odA == bdA && oiA < biA)) { bdA = odA; biA = oiA; }
      if (odB < bdB || (odB == bdB && oiB < biB)) { bdB = odB; biB = oiB; }
    }
    if (l15 == 0) {
      const int mA = m0 + j + 8 * half;
      const int mB = m0 + 16 + j + 8 * half;
      bdp[(size_t)split * NUM_ROWS + mA] = bdA;  bip[(size_t)split * NUM_ROWS + mA] = biA;
      bdp[(size_t)split * NUM_ROWS + mB] = bdB;  bip[(size_t)split * NUM_ROWS + mB] = biB;
    }
  }
}

// ---------------------------------------------------------------------------
// Combine per-split candidates into the final argmin per row.
// Splits visited in increasing index order; strict < preserves exact
// first-minimum (argmin) semantics.
// ---------------------------------------------------------------------------
__global__ __launch_bounds__(256) void k_select(const float* __restrict__ bdp,
                                                const int* __restrict__ bip,
                                                float* __restrict__ idxf,
                                                int* __restrict__ idxi) {
  const int m = blockIdx.x * 256 + threadIdx.x;
  float bd = bdp[m];
  int   bi = bip[m];
#pragma unroll
  for (int s = 1; s < NSPLIT; ++s) {
    const float d = bdp[(size_t)s * NUM_ROWS + m];
    const int   i = bip[(size_t)s * NUM_ROWS + m];
    if (d < bd || (d == bd && i < bi)) { bd = d; bi = i; }
  }
  idxf[m] = (float)bi;
  idxi[m] = bi;
}

// ---------------------------------------------------------------------------
// hard_ste = book[idx] (STE forward value == hard), plus per-block partial
// sums of (hard - soft)^2. Element-parallel, coalesced.
// ---------------------------------------------------------------------------
__global__ __launch_bounds__(256) void k_gather(const float* __restrict__ soft,
                                                const float* __restrict__ book,
                                                const int* __restrict__ idxi,
                                                float* __restrict__ hard,
                                                float* __restrict__ partial) {
  __shared__ float sdata[256];
  const int e   = blockIdx.x * 256 + threadIdx.x;
  const int row = e >> 6;
  const int c   = e & 63;
  const int bi  = idxi[row];
  const float hv = book[(size_t)bi * CODE_DEPTH + c];
  const float sv = soft[(size_t)e];
  hard[(size_t)e] = hv;
  const float d = hv - sv;
  sdata[threadIdx.x] = d * d;
  __syncthreads();
#pragma unroll
  for (int st = 128; st > 0; st >>= 1) {
    if ((int)threadIdx.x < st) sdata[threadIdx.x] += sdata[threadIdx.x + st];
    __syncthreads();
  }
  if (threadIdx.x == 0) partial[blockIdx.x] = sdata[0];
}

// ---------------------------------------------------------------------------
// Final deterministic reduction; both losses equal in the forward pass.
// ---------------------------------------------------------------------------
__global__ __launch_bounds__(256) void k_loss(const float* __restrict__ partial,
                                              float* __restrict__ out_c,
                                              float* __restrict__ out_e) {
  __shared__ float sdata[256];
  float s = 0.f;
  for (int i = threadIdx.x; i < 4096; i += 256) s += partial[i];
  sdata[threadIdx.x] = s;
  __syncthreads();
#pragma unroll
  for (int st = 128; st > 0; st >>= 1) {
    if ((int)threadIdx.x < st) sdata[threadIdx.x] += sdata[threadIdx.x + st];
    __syncthreads();
  }
  if (threadIdx.x == 0) {
    const float mean = sdata[0] / (float)((size_t)NUM_ROWS * CODE_DEPTH);
    *out_c = mean;
    *out_e = mean;
  }
}

extern "C" void kernel_launch(void* const* d_in, const int* in_sizes, int n_in,
                              void* d_out, int out_size, void* d_ws, size_t ws_size,
                              hipStream_t stream) {
  const float* x    = (const float*)d_in[0];   // [8,2048,1024]
  const float* W    = (const float*)d_in[1];   // [64,1024]
  const float* bias = (const float*)d_in[2];   // [64]
  const float* book = (const float*)d_in[3];   // [1,8192,64]

  float* out  = (float*)d_out;
  float* soft = out;                                   // 1,048,576
  float* idxf = out + (size_t)NUM_ROWS * CODE_DEPTH;   // 16,384
  float* hard = idxf + NUM_ROWS;                       // 1,048,576
  float* closs = hard + (size_t)NUM_ROWS * CODE_DEPTH; // 1
  float* eloss = closs + 1;                            // 1

  char* ws = (char*)d_ws;
  float* c2      = (float*)ws;                          ws += (size_t)NUM_CODES * 4;           // 8192 f
  int*   idxi    = (int*)ws;                            ws += (size_t)NUM_ROWS * 4;            // 16384 i
  float* partial = (float*)ws;                          ws += (size_t)4096 * 4;                // 4096 f
  float* bdp     = (float*)ws;                          ws += (size_t)NSPLIT * NUM_ROWS * 4;   // 131072 f
  int*   bip     = (int*)ws;                            /* 131072 i */

  k_down  <<<NUM_ROWS / 128, 256, 0, stream>>>(x, W, bias, soft);
  k_c2    <<<NUM_CODES / 256, 256, 0, stream>>>(book, c2);
  k_argmin<<<(NUM_ROWS / 256) * NSPLIT, 256, 0, stream>>>(soft, book, c2, bdp, bip);
  k_select<<<NUM_ROWS / 256, 256, 0, stream>>>(bdp, bip, idxf, idxi);
  k_gather<<<(NUM_ROWS * CODE_DEPTH) / 256, 256, 0, stream>>>(soft, book, idxi, hard, partial);
  k_loss  <<<1, 256, 0, stream>>>(partial, closs, eloss);
}